// ExtractFeature_6425271074845
// MI455X (gfx1250) — compile-verified
//
#include <hip/hip_runtime.h>

#define VOCAB 50000
#define EMB   300
#define HID   150
#define BB    32
#define SS    128
#define BS    (BB * SS)     // 4096 rows
#define HPAD  160           // HID padded to multiple of 16
#define NPAD  (2 * HPAD)    // fused [Wq|Wk] output width = 320
#define BKPAD 308           // LDS row pad for conflict-free ds_load_b64
#define BCOLS 64            // N columns staged per block (4 tiles of 16)

typedef __attribute__((ext_vector_type(2))) float v2f;
typedef __attribute__((ext_vector_type(8))) float v8f;

#if defined(__HIP_DEVICE_COMPILE__) && __has_builtin(__builtin_amdgcn_global_load_async_to_lds_b32)
#define HAVE_ASYNC_LDS 1
typedef __attribute__((address_space(1))) int gint_t;
typedef __attribute__((address_space(3))) int lint_t;
#else
#define HAVE_ASYNC_LDS 0
#endif

// ---------------------------------------------------------------------------
// Workspace layout (floats):
//   Wpad    : [EMB][NPAD]        = 96,000
//   biaspad : [NPAD]             = 320
//   qkp     : [BS][NPAD]         = 1,310,720   (qp in cols [0,160), kp in [160,320))
//   wacc    : [BB][SS]           = 4,096       (softmax column weights, mean over q)
// total ~5.65 MB
// ---------------------------------------------------------------------------

__global__ void pack_weights_kernel(const float* __restrict__ Wq,
                                    const float* __restrict__ Wk,
                                    const float* __restrict__ bq,
                                    const float* __restrict__ bk,
                                    float* __restrict__ Wpad,
                                    float* __restrict__ biaspad,
                                    float* __restrict__ wacc) {
    int idx = blockIdx.x * blockDim.x + threadIdx.x;
    if (idx < EMB * NPAD) {
        int k = idx / NPAD;
        int n = idx - k * NPAD;
        float v = 0.0f;
        if (n < HPAD) {
            if (n < HID) v = Wq[k * HID + n];
        } else {
            int c = n - HPAD;
            if (c < HID) v = Wk[k * HID + c];
        }
        Wpad[idx] = v;
    }
    if (idx < NPAD) {
        float v = 0.0f;
        if (idx < HPAD) {
            if (idx < HID) v = bq[idx];
        } else {
            int c = idx - HPAD;
            if (c < HID) v = bk[c];
        }
        biaspad[idx] = v;
    }
    if (idx < BS) wacc[idx] = 0.0f;   // zero the softmax-weight accumulator
}

// Embedding gather: emb[row, :] = table[X[row], :]   (row = b*S + s)
// Rows are 1200 B, 8-byte aligned -> float2 copies.
__global__ void gather_emb_kernel(const int* __restrict__ X,
                                  const float* __restrict__ table,
                                  float* __restrict__ emb_out) {
    int row = blockIdx.x;
    long long tok = (long long)X[row];
    const v2f* src = (const v2f*)(table + tok * EMB);
    v2f* dst = (v2f*)(emb_out + (long long)row * EMB);
    for (int e = threadIdx.x; e < EMB / 2; e += blockDim.x) dst[e] = src[e];
}

// Fused Q/K projection GEMM via V_WMMA_F32_16X16X4_F32:
//   qkp[4096, 320] = emb[4096, 300] x Wpad[300, 320] + biaspad
// Each wave computes 4 adjacent 16x16 C tiles (one A fragment feeds 4 WMMAs);
// 8 waves per block share a 300x64 B tile staged once into LDS (transposed,
// row stride 308 words -> every fragment's {k,k+1} b64 read hits all 64 banks
// exactly once; fragment row offset 16*308 = 0 mod 64 preserves the pattern),
// staged via CDNA5 async global->LDS scatter (ASYNCcnt).
__global__ void __launch_bounds__(256)
qk_gemm_wmma_kernel(const float* __restrict__ emb,
                    const float* __restrict__ Wpad,
                    const float* __restrict__ biaspad,
                    float* __restrict__ qkp) {
    __shared__ float sBT[BCOLS][BKPAD];       // [n][k]

    const int tid   = threadIdx.x;
    const int lane  = tid & 31;
    const int wave  = tid >> 5;
    const int Nbase = blockIdx.x * BCOLS;
    const int Mbase = (blockIdx.y * 8 + wave) * 16;

    // ---- stage B tile: sBT[n][k] = Wpad[k][Nbase+n], n = 0..63, k = 0..299 ----
    {
        const int n  = tid & (BCOLS - 1);
        const int k0 = tid >> 6;              // 0..3
#if HAVE_ASYNC_LDS
        for (int k = k0; k < EMB; k += 4) {
            __builtin_amdgcn_global_load_async_to_lds_b32(
                (gint_t*)(Wpad + (long long)k * NPAD + Nbase + n),
                (lint_t*)&sBT[n][k],
                /*offset=*/0, /*cpol=*/0);
        }
#if __has_builtin(__builtin_amdgcn_s_wait_asynccnt)
        __builtin_amdgcn_s_wait_asynccnt(0);
#else
        asm volatile("s_wait_asynccnt 0x0" ::: "memory");
#endif
#else
        for (int k = k0; k < EMB; k += 4)
            sBT[n][k] = Wpad[(long long)k * NPAD + Nbase + n];
#endif
    }
    __syncthreads();

    // ISA 16x4 f32 A layout: lanes 0-15 carry K={kb,kb+1} for M=lane,
    //                        lanes 16-31 carry K={kb+2,kb+3} for M=lane-16.
    const int nloc  = lane & 15;
    const int khalf = (lane >> 4) * 2;        // 0 or 2

    const float* arow  = emb + (long long)(Mbase + nloc) * EMB + khalf;
    const float* brow  = &sBT[nloc][khalf];

    v8f c0 = {}, c1 = {}, c2 = {}, c3 = {};
    #pragma unroll 5
    for (int kb = 0; kb < EMB; kb += 4) {
        v2f a  = *(const v2f*)(arow);                     // global b64
        v2f b0 = *(const v2f*)(brow);                     // ds b64, tile n+0
        v2f b1 = *(const v2f*)(brow + 16 * BKPAD);        // tile n+16
        v2f b2 = *(const v2f*)(brow + 32 * BKPAD);        // tile n+32
        v2f b3 = *(const v2f*)(brow + 48 * BKPAD);        // tile n+48
        c0 = __builtin_amdgcn_wmma_f32_16x16x4_f32(false, a, false, b0, (short)0, c0, false, false);
        c1 = __builtin_amdgcn_wmma_f32_16x16x4_f32(false, a, false, b1, (short)0, c1, false, false);
        c2 = __builtin_amdgcn_wmma_f32_16x16x4_f32(false, a, false, b2, (short)0, c2, false, false);
        c3 = __builtin_amdgcn_wmma_f32_16x16x4_f32(false, a, false, b3, (short)0, c3, false, false);
        arow += 4;
        brow += 4;
    }

    // C layout: VGPR i, lanes 0-15 -> (M = Mbase+i,   N = Ntile+lane)
    //                  lanes 16-31 -> (M = Mbase+8+i, N = Ntile+lane-16)
    const int mtop = Mbase + ((lane & 16) ? 8 : 0);
    {
        const float bias = biaspad[Nbase + nloc];
        #pragma unroll
        for (int i = 0; i < 8; ++i)
            qkp[(long long)(mtop + i) * NPAD + Nbase + nloc] = c0[i] + bias;
    }
    {
        const float bias = biaspad[Nbase + 16 + nloc];
        #pragma unroll
        for (int i = 0; i < 8; ++i)
            qkp[(long long)(mtop + i) * NPAD + Nbase + 16 + nloc] = c1[i] + bias;
    }
    {
        const float bias = biaspad[Nbase + 32 + nloc];
        #pragma unroll
        for (int i = 0; i < 8; ++i)
            qkp[(long long)(mtop + i) * NPAD + Nbase + 32 + nloc] = c2[i] + bias;
    }
    {
        const float bias = biaspad[Nbase + 48 + nloc];
        #pragma unroll
        for (int i = 0; i < 8; ++i)
            qkp[(long long)(mtop + i) * NPAD + Nbase + 48 + nloc] = c3[i] + bias;
    }
}

// Fused additive-attention scores + softmax + column-weight accumulation.
// One block per (b, q) row; lane t owns key k = t.
// w[b,k] += softmax_k( sum_h tanh(qp[b,q,h]+kp[b,k,h]) * wv[h] + bv ) / S
__global__ void __launch_bounds__(128)
attn_kernel(const float* __restrict__ qkp,
            const float* __restrict__ wv,
            const float* __restrict__ bv,
            float* __restrict__ wacc) {
    __shared__ float qs[HID];
    __shared__ float wvs[HID];
    __shared__ float red[SS];

    const int t   = threadIdx.x;        // key index 0..127
    const int row = blockIdx.x;         // b*S + q
    const int b   = row >> 7;

    for (int h = t; h < HID; h += 128) {
        qs[h]  = qkp[(long long)row * NPAD + h];
        wvs[h] = wv[h];
    }
    __syncthreads();

    const float* kprow = qkp + (long long)(b * SS + t) * NPAD + HPAD;
    float score = bv[0];
    #pragma unroll 5
    for (int h = 0; h < HID; h += 2) {
        v2f kp2 = *(const v2f*)(kprow + h);   // 8B aligned (HPAD even, h even)
        score += tanhf(qs[h]     + kp2.x) * wvs[h];
        score += tanhf(qs[h + 1] + kp2.y) * wvs[h + 1];
    }

    // softmax over the 128 keys (4 waves) via LDS tree reductions
    red[t] = score;
    __syncthreads();
    for (int off = 64; off > 0; off >>= 1) {
        if (t < off) red[t] = fmaxf(red[t], red[t + off]);
        __syncthreads();
    }
    const float m = red[0];
    __syncthreads();
    const float e = __expf(score - m);
    red[t] = e;
    __syncthreads();
    for (int off = 64; off > 0; off >>= 1) {
        if (t < off) red[t] += red[t + off];
        __syncthreads();
    }
    const float a = e / red[0];

    // mean over q folded in: accumulate attn/S
    atomicAdd(&wacc[b * SS + t], a * (1.0f / (float)SS));
}

// pooled[b, e] = sum_k wacc[b,k] * emb[b,k,e]
__global__ void __launch_bounds__(320)
pooled_kernel(const float* __restrict__ emb,
              const float* __restrict__ wacc,
              float* __restrict__ pooled) {
    const int b = blockIdx.x;
    const int e = threadIdx.x;
    if (e >= EMB) return;
    const float* wrow = wacc + b * SS;
    const float* embb = emb + (long long)b * SS * EMB;
    float acc = 0.0f;
    for (int k = 0; k < SS; ++k) {
        acc += wrow[k] * embb[(long long)k * EMB + e];
    }
    pooled[b * EMB + e] = acc;
}

extern "C" void kernel_launch(void* const* d_in, const int* in_sizes, int n_in,
                              void* d_out, int out_size, void* d_ws, size_t ws_size,
                              hipStream_t stream) {
    (void)in_sizes; (void)n_in; (void)out_size; (void)ws_size;

    const int*   X     = (const int*)d_in[0];
    const float* table = (const float*)d_in[1];
    const float* Wq    = (const float*)d_in[2];
    const float* bq    = (const float*)d_in[3];
    const float* Wk    = (const float*)d_in[4];
    const float* bk    = (const float*)d_in[5];
    const float* wv    = (const float*)d_in[6];
    const float* bv    = (const float*)d_in[7];

    float* out    = (float*)d_out;
    float* emb    = out;                              // output 0: [4096][300]
    float* pooled = out + (long long)BS * EMB;        // output 1: [32][300]

    float* ws      = (float*)d_ws;
    float* Wpad    = ws;                              // [300][320]
    float* biaspad = Wpad + EMB * NPAD;               // [320]
    float* qkp     = biaspad + NPAD;                  // [4096][320]
    float* wacc    = qkp + (long long)BS * NPAD;      // [32][128]

    pack_weights_kernel<<<(EMB * NPAD + 255) / 256, 256, 0, stream>>>(
        Wq, Wk, bq, bk, Wpad, biaspad, wacc);

    gather_emb_kernel<<<BS, 128, 0, stream>>>(X, table, emb);

    dim3 gGemm(NPAD / BCOLS, BS / (16 * 8));          // (5, 32)
    qk_gemm_wmma_kernel<<<gGemm, 256, 0, stream>>>(emb, Wpad, biaspad, qkp);

    attn_kernel<<<BS, 128, 0, stream>>>(qkp, wv, bv, wacc);

    pooled_kernel<<<BB, 320, 0, stream>>>(emb, wacc, pooled);
}